// ConvLSTM_39642548142821
// MI455X (gfx1250) — compile-verified
//
#include <hip/hip_runtime.h>

// Problem constants (from reference)
#define TT   25
#define BB   16
#define CIN  3
#define HID  32
#define WW   4096
#define OCH  128          // 4*HID
#define TW   64           // W-tile per block
#define NT4  (TW/16)      // 4 N-tiles of 16 per wave
#define PSTR 66           // padded row stride for [ch][TW] LDS arrays (bank-friendly)
#define CSTRC 34          // col stride for cell-state tile [col][HID+2]

typedef __attribute__((ext_vector_type(2))) float v2f;
typedef __attribute__((ext_vector_type(8))) float v8f;

__device__ __forceinline__ float sigm_(float x) { return 1.0f / (1.0f + __expf(-x)); }

__device__ __forceinline__ v8f wmma_f32(v2f a, v2f b, v8f c) {
  // D = A(16x4) * B(4x16) + C   (fp32 WMMA, CDNA5)
  return __builtin_amdgcn_wmma_f32_16x16x4_f32(
      /*neg_a=*/false, a, /*neg_b=*/false, b,
      /*c_mod=*/(short)0, c, /*reuse_a=*/false, /*reuse_b=*/false);
}

// ---- One-time conv-weight repack: cw (O, I, K) -> wre (K, O, CPAD), zero-padded.
// Makes every WMMA A-fragment one aligned 8-byte global load with no guards.
template <int CIN_TOT, int CPAD>
__global__ __launch_bounds__(256)
void repack_w(const float* __restrict__ cw, float* __restrict__ wre) {
  const int idx = blockIdx.x * 256 + threadIdx.x;
  if (idx >= 3 * OCH * CPAD) return;
  const int i = idx % CPAD;
  const int o = (idx / CPAD) % OCH;
  const int k = idx / (CPAD * OCH);
  wre[idx] = (i < CIN_TOT) ? cw[((long)o * CIN_TOT + i) * 3 + k] : 0.0f;
}

// One ConvLSTM cell step for one layer over the whole (B, W) domain.
// CIN_TOT = conv input channels (x-channels + HID).  CPAD = CIN_TOT padded to x4.
template <int CIN_TOT, int CPAD>
__global__ __launch_bounds__(256)
void convlstm_step(const float* __restrict__ xsrc, long x_bstride,   // x slice for this t; ch stride = WW
                   const float* __restrict__ hprev, long h_bstride,  // h(t-1) slice or nullptr at t==0
                   const float* __restrict__ wre,   // repacked (3, OCH, CPAD)
                   const float* __restrict__ cb,    // (OCH)
                   const float* __restrict__ wci, const float* __restrict__ wcf,
                   const float* __restrict__ wco,   // (HID, HID)
                   float* __restrict__ cstate,      // (B, HID, W) cell state (in/out)
                   float* __restrict__ yout, long y_bstride,  // h(t) destination slice
                   int t)
{
  constexpr int CSTR = CPAD + 2;               // col stride, (CSTR/2) odd -> 64-bank clean
  extern __shared__ float smem[];
  float* in_s = smem;                          // [TW+2][CSTR]  column-major halo'd input
  float* c_s  = in_s + (TW + 2) * CSTR;        // [TW][CSTRC]   column-major cell state
  float* cc_s = c_s + TW * CSTRC;              // [OCH][PSTR]   conv result (i,f,o,g)
  float* pi_s = cc_s + OCH * PSTR;             // [HID][PSTR]   peephole wci*c
  float* pf_s = pi_s + HID * PSTR;             // [HID][PSTR]   peephole wcf*c
  float* po_s = pf_s + HID * PSTR;             // [HID][PSTR]   peephole wco*c_next

  const int tid  = threadIdx.x;
  const int lane = tid & 31;
  const int wv   = tid >> 5;                   // 0..7 (8 waves)
  const int b      = blockIdx.x / (WW / TW);
  const int w_base = (blockIdx.x % (WW / TW)) * TW;

  constexpr int CIN_X = CIN_TOT - HID;         // x-channel count

  // ---------------- Phase 1: stage inputs into LDS ----------------
  // Global reads stay fully coalesced (col fastest); LDS writes are strided
  // by CSTR which is conflict-free across 32 lanes.
  for (int idx = tid; idx < CPAD * (TW + 2); idx += 256) {
    const int ch  = idx / (TW + 2);
    const int col = idx % (TW + 2);
    const int wg  = w_base - 1 + col;          // col 0 == global w_base-1 (SAME pad)
    float v = 0.0f;
    if (wg >= 0 && wg < WW) {
      if (ch < CIN_X) {
        v = xsrc[(long)b * x_bstride + (long)ch * WW + wg];
      } else if (ch < CIN_TOT) {
        if (hprev) v = hprev[(long)b * h_bstride + (long)(ch - CIN_X) * WW + wg];
      }
    }
    in_s[col * CSTR + ch] = v;
  }
  for (int idx = tid; idx < HID * TW; idx += 256) {
    const int ch = idx / TW, col = idx % TW;
    c_s[col * CSTRC + ch] =
        (t > 0) ? cstate[((long)b * HID + ch) * WW + (w_base + col)] : 0.0f;
  }
  __syncthreads();

  const int m      = lane & 15;                // M (A rows) / N (B,C cols)
  const int kb     = (lane >> 4) * 2;          // K base for A/B fragments
  const int rowoff = (lane >> 4) * 8;          // M offset for C/D rows

  // ---------------- Phase 2a: conv as WMMA f32 16x16x4 ----------------
  {
    const int o16 = wv * 16;                   // this wave's output-channel tile
    v8f acc[NT4];
#pragma unroll
    for (int n = 0; n < NT4; ++n) acc[n] = {};
#pragma unroll
    for (int k = 0; k < 3; ++k) {
      const float* wrow = wre + ((long)k * OCH + o16 + m) * CPAD;
      for (int c4 = 0; c4 < CPAD / 4; ++c4) {
        const int i0 = c4 * 4 + kb;
        const v2f a = *(const v2f*)(wrow + i0);          // one global b64
#pragma unroll
        for (int n = 0; n < NT4; ++n) {
          const int col = n * 16 + m + k;                // w_local + k in halo'd tile
          const v2f bf = *(const v2f*)(in_s + col * CSTR + i0);  // one ds b64
          acc[n] = wmma_f32(a, bf, acc[n]);
        }
      }
    }
#pragma unroll
    for (int n = 0; n < NT4; ++n)
#pragma unroll
      for (int r = 0; r < 8; ++r) {
        const int oc = o16 + r + rowoff;
        cc_s[oc * PSTR + n * 16 + m] = acc[n][r] + cb[oc];
      }
  }

  // ---------------- Phase 2b: peephole wci*c and wcf*c (WMMA) ----------------
  {
    const int mt = wv >> 2, nt = wv & 3;       // 2 M-tiles x 4 N-tiles over (HID, TW)
    const int o16 = mt * 16;
    v8f pacc = {}, facc = {};
    for (int c4 = 0; c4 < HID / 4; ++c4) {
      const int i0 = c4 * 4 + kb;
      const v2f ai = *(const v2f*)(wci + (o16 + m) * HID + i0);
      const v2f af = *(const v2f*)(wcf + (o16 + m) * HID + i0);
      const v2f bf = *(const v2f*)(c_s + (nt * 16 + m) * CSTRC + i0);
      pacc = wmma_f32(ai, bf, pacc);
      facc = wmma_f32(af, bf, facc);
    }
#pragma unroll
    for (int r = 0; r < 8; ++r) {
      const int oc = o16 + r + rowoff;
      pi_s[oc * PSTR + nt * 16 + m] = pacc[r];
      pf_s[oc * PSTR + nt * 16 + m] = facc[r];
    }
  }
  __syncthreads();

  // ---------------- Phase 3: gates, c update ----------------
  for (int idx = tid; idx < HID * TW; idx += 256) {
    const int o = idx / TW, wl = idx % TW;
    const float co = c_s[wl * CSTRC + o];
    const float iv = sigm_(cc_s[o * PSTR + wl] + pi_s[o * PSTR + wl]);            // cc_i
    const float fv = sigm_(cc_s[(HID + o) * PSTR + wl] + pf_s[o * PSTR + wl]);    // cc_f
    const float gv = tanhf(cc_s[(3 * HID + o) * PSTR + wl]);                      // cc_g
    const float cn = fv * co + iv * gv;
    c_s[wl * CSTRC + o] = cn;
    cstate[((long)b * HID + o) * WW + (w_base + wl)] = cn;
  }
  __syncthreads();

  // ---------------- Phase 4: peephole wco*c_next (WMMA) ----------------
  {
    const int mt = wv >> 2, nt = wv & 3;
    const int o16 = mt * 16;
    v8f oacc = {};
    for (int c4 = 0; c4 < HID / 4; ++c4) {
      const int i0 = c4 * 4 + kb;
      const v2f a  = *(const v2f*)(wco + (o16 + m) * HID + i0);
      const v2f bf = *(const v2f*)(c_s + (nt * 16 + m) * CSTRC + i0);
      oacc = wmma_f32(a, bf, oacc);
    }
#pragma unroll
    for (int r = 0; r < 8; ++r)
      po_s[(o16 + r + rowoff) * PSTR + nt * 16 + m] = oacc[r];
  }
  __syncthreads();

  // ---------------- Phase 5: h output ----------------
  for (int idx = tid; idx < HID * TW; idx += 256) {
    const int o = idx / TW, wl = idx % TW;
    const float cn = c_s[wl * CSTRC + o];
    const float ov = sigm_(cc_s[(2 * HID + o) * PSTR + wl] + po_s[o * PSTR + wl]); // cc_o
    yout[(long)b * y_bstride + (long)o * WW + (w_base + wl)] = ov * tanhf(cn);
  }
}

extern "C" void kernel_launch(void* const* d_in, const int* in_sizes, int n_in,
                              void* d_out, int out_size, void* d_ws, size_t ws_size,
                              hipStream_t stream) {
  const float* x    = (const float*)d_in[0];
  const float* cw0  = (const float*)d_in[1];
  const float* cb0  = (const float*)d_in[2];
  const float* wci0 = (const float*)d_in[3];
  const float* wcf0 = (const float*)d_in[4];
  const float* wco0 = (const float*)d_in[5];
  const float* cw1  = (const float*)d_in[6];
  const float* cb1  = (const float*)d_in[7];
  const float* wci1 = (const float*)d_in[8];
  const float* wcf1 = (const float*)d_in[9];
  const float* wco1 = (const float*)d_in[10];
  float* out = (float*)d_out;                      // (B, T, HID, W)

  // Workspace: y0 (T,B,HID,W), two cell states, repacked conv weights.
  float* y0   = (float*)d_ws;
  float* c0   = y0 + (size_t)TT * BB * HID * WW;
  float* c1   = c0 + (size_t)BB * HID * WW;
  float* wre0 = c1 + (size_t)BB * HID * WW;        // (3, OCH, 36)
  float* wre1 = wre0 + (size_t)3 * OCH * 36;       // (3, OCH, 64)

  repack_w<35, 36><<<(3 * OCH * 36 + 255) / 256, 256, 0, stream>>>(cw0, wre0);
  repack_w<64, 64><<<(3 * OCH * 64 + 255) / 256, 256, 0, stream>>>(cw1, wre1);

  const dim3 grid(BB * (WW / TW));
  const dim3 blk(256);
  const size_t lds_common =
      (size_t)(TW * CSTRC + OCH * PSTR + 3 * HID * PSTR) * sizeof(float);
  const size_t lds0 = (size_t)((TW + 2) * (36 + 2)) * sizeof(float) + lds_common;
  const size_t lds1 = (size_t)((TW + 2) * (64 + 2)) * sizeof(float) + lds_common;

  for (int t = 0; t < TT; ++t) {
    // Layer 0: x channels=3, hidden=32 -> CIN_TOT=35, pad to 36.
    convlstm_step<35, 36><<<grid, blk, lds0, stream>>>(
        x + (size_t)t * CIN * WW, (long)TT * CIN * WW,
        t ? y0 + (size_t)(t - 1) * BB * HID * WW : nullptr, (long)HID * WW,
        wre0, cb0, wci0, wcf0, wco0, c0,
        y0 + (size_t)t * BB * HID * WW, (long)HID * WW, t);
    // Layer 1: x channels=32 (y0), hidden=32 -> CIN_TOT=64.
    convlstm_step<64, 64><<<grid, blk, lds1, stream>>>(
        y0 + (size_t)t * BB * HID * WW, (long)HID * WW,
        t ? out + (size_t)(t - 1) * HID * WW : nullptr, (long)TT * HID * WW,
        wre1, cb1, wci1, wcf1, wco1, c1,
        out + (size_t)t * HID * WW, (long)TT * HID * WW, t);
  }
}